// GCNLayer_74036646248900
// MI455X (gfx1250) — compile-verified
//
#include <hip/hip_runtime.h>

// ---------------------------------------------------------------------------
// GCNConv on MI455X (gfx1250, wave32).
//   out = D^{-1/2} (A + I) D^{-1/2} (x @ W) + b
// Dense transform via V_WMMA_F32_16X16X4_F32 (full f32 precision); one wave
// computes a 16x128 row stripe (8 accumulators) so each A fragment is loaded
// once and feeds 8 independent WMMAs. Sparse aggregation via wave-per-edge
// f32 atomics (the memory/L2-bound part).
// ---------------------------------------------------------------------------

typedef __attribute__((ext_vector_type(2))) float v2f;
typedef __attribute__((ext_vector_type(8))) float v8f;

#define DFEAT 128

// --- 1) zero output (float4), init degree to 1.0 (self loop) ----------------
__global__ void k_init(float4* __restrict__ out4, float* __restrict__ deg,
                       int n_out4, int n_nodes) {
  int i = blockIdx.x * blockDim.x + threadIdx.x;
  if (i < n_out4) out4[i] = make_float4(0.f, 0.f, 0.f, 0.f);
  if (i < n_nodes) deg[i] = 1.0f;
}

// --- 2) degree accumulation over dst ----------------------------------------
__global__ void k_deg(const int* __restrict__ dst, float* __restrict__ deg, int E) {
  int e = blockIdx.x * blockDim.x + threadIdx.x;
  if (e < E) atomicAdd(&deg[dst[e]], 1.0f);
}

// --- 3) deg -> rsqrt(deg) in place (deg >= 1 always) ------------------------
__global__ void k_dinv(float* __restrict__ deg, int n) {
  int i = blockIdx.x * blockDim.x + threadIdx.x;
  if (i < n) deg[i] = rsqrtf(deg[i]);
}

// --- 4) xw = x @ W via WMMA f32 16x16x4 -------------------------------------
// One wave per 16-row stripe; 8 accumulators cover N = 0..127.
// Per ISA 7.12.2 (32-bit A 16x4): lanes 0-15 hold K={0,1}, lanes 16-31 K={2,3},
// row M = lane%16. B (4x16) mirrors: VGPR0 = Krow {0|2}, VGPR1 = Krow {1|3},
// col N = lane%16. C/D: VGPR i holds rows i (lanes 0-15) and i+8 (lanes 16-31).
__global__ void k_gemm(const float* __restrict__ x, const float* __restrict__ W,
                       float* __restrict__ xw, int n_nodes) {
  const int lane = threadIdx.x & 31;
  const int wave = threadIdx.x >> 5;
  const int half = lane >> 4;   // 0 or 1
  const int r    = lane & 15;

  const int stripe = blockIdx.x * 8 + wave;   // 16-row stripe index
  int m0 = stripe * 16;
  if (m0 >= n_nodes) return;                  // wave-uniform exit (EXEC stays full)
  if (m0 + 16 > n_nodes) m0 = n_nodes - 16;   // clamp: duplicate tile, same values

  v8f c[8];
#pragma unroll
  for (int t = 0; t < 8; ++t) c[t] = (v8f){};

  const float* __restrict__ arow = x + (size_t)(m0 + r) * DFEAT;

#pragma unroll 2
  for (int k = 0; k < DFEAT; k += 4) {
    // A: row M = r, contiguous K pair (k+2*half, k+2*half+1) -> one b64 load
    const v2f a = *(const v2f*)(arow + k + 2 * half);
    const float* __restrict__ w0 = W + (size_t)(k + 2 * half) * DFEAT + r;
    const float* __restrict__ w1 = w0 + DFEAT;
#pragma unroll
    for (int t = 0; t < 8; ++t) {
      v2f b;
      b.x = w0[t * 16];
      b.y = w1[t * 16];
      // 8 args: (neg_a, A, neg_b, B, c_mod, C, reuse_a, reuse_b)
      c[t] = __builtin_amdgcn_wmma_f32_16x16x4_f32(false, a, false, b,
                                                   (short)0, c[t], false, false);
    }
  }

#pragma unroll
  for (int t = 0; t < 8; ++t) {
#pragma unroll
    for (int i = 0; i < 8; ++i) {
      xw[(size_t)(m0 + i + 8 * half) * DFEAT + t * 16 + r] = c[t][i];
    }
  }
}

// --- 5) edge scatter: one wave32 per edge -----------------------------------
// norm = dinv[src]*dinv[dst]; out[dst,:] += norm * xw[src,:]
// Each lane: one float4 of the 128-wide row (coalesced 512B gather from L2),
// then 4 global f32 atomics.
__global__ void k_scatter(const int* __restrict__ src, const int* __restrict__ dst,
                          const float* __restrict__ dinv,
                          const float* __restrict__ xw,
                          float* __restrict__ out, int E) {
  const int lane = threadIdx.x & 31;
  const int wg = (blockIdx.x * blockDim.x + threadIdx.x) >> 5;  // global wave id
  if (wg >= E) return;

  const int s = src[wg];
  const int d = dst[wg];
  const float norm = dinv[s] * dinv[d];

  const float4 v = ((const float4*)(xw + (size_t)s * DFEAT))[lane];
  float* __restrict__ od = out + (size_t)d * DFEAT + lane * 4;
  atomicAdd(od + 0, norm * v.x);
  atomicAdd(od + 1, norm * v.y);
  atomicAdd(od + 2, norm * v.z);
  atomicAdd(od + 3, norm * v.w);
}

// --- 6) self-loop term + bias (float4) --------------------------------------
// out[i,:] += dinv[i]^2 * xw[i,:] + b[:]; one thread per float4, runs after
// all edge atomics on the same stream, so a plain RMW is safe.
__global__ void k_final(const float4* __restrict__ xw4, const float* __restrict__ dinv,
                        const float4* __restrict__ b4, float4* __restrict__ out4,
                        int total4) {
  int idx = blockIdx.x * blockDim.x + threadIdx.x;
  if (idx >= total4) return;
  const int node = idx >> 5;      // (idx*4) / 128
  const int ch4  = idx & 31;      // float4 index within the 128-wide row
  const float di = dinv[node];
  const float s  = di * di;
  const float4 xv = xw4[idx];
  const float4 bv = b4[ch4];
  float4 ov = out4[idx];
  ov.x += s * xv.x + bv.x;
  ov.y += s * xv.y + bv.y;
  ov.z += s * xv.z + bv.z;
  ov.w += s * xv.w + bv.w;
  out4[idx] = ov;
}

// ---------------------------------------------------------------------------
extern "C" void kernel_launch(void* const* d_in, const int* in_sizes, int n_in,
                              void* d_out, int out_size, void* d_ws, size_t ws_size,
                              hipStream_t stream) {
  const float* x  = (const float*)d_in[0];
  const int*   ei = (const int*)d_in[1];     // [2, E] row-major
  const float* W  = (const float*)d_in[2];   // [128, 128]
  const float* b  = (const float*)d_in[3];   // [128]
  float* out = (float*)d_out;

  const int n_nodes = in_sizes[0] / DFEAT;   // 100000
  const int E       = in_sizes[1] / 2;       // 600000
  const int* src = ei;
  const int* dst = ei + E;

  // workspace: xw [n_nodes*128 f32] then deg/dinv [n_nodes f32]
  float* xw  = (float*)d_ws;
  float* deg = xw + (size_t)n_nodes * DFEAT;

  const int total  = n_nodes * DFEAT;
  const int total4 = total / 4;

  k_init<<<(total4 + 255) / 256, 256, 0, stream>>>((float4*)out, deg, total4, n_nodes);
  k_deg<<<(E + 255) / 256, 256, 0, stream>>>(dst, deg, E);
  k_dinv<<<(n_nodes + 255) / 256, 256, 0, stream>>>(deg, n_nodes);

  // one wave per 16-row stripe; 8 waves (256 threads) per block
  const int stripes = (n_nodes + 15) / 16;         // 6250
  k_gemm<<<(stripes + 7) / 8, 256, 0, stream>>>(x, W, xw, n_nodes);

  // one wave (32 lanes) per edge; 8 waves per 256-thread block
  k_scatter<<<(E + 7) / 8, 256, 0, stream>>>(src, dst, deg, xw, out, E);

  k_final<<<(total4 + 255) / 256, 256, 0, stream>>>((const float4*)xw, deg,
                                                    (const float4*)b, (float4*)out,
                                                    total4);
}